// TorchTopologicalPersistenceLoss_25666724561259
// MI455X (gfx1250) — compile-verified
//
#include <hip/hip_runtime.h>
#include <math.h>

typedef float v2f __attribute__((ext_vector_type(2)));
typedef float v8f __attribute__((ext_vector_type(8)));

#define N_CLS 3
#define NP    64
#define DIMF  768
#define NE    2016      // C(64,2)
#define NT    41664     // C(64,3)
#define NSORT 65536     // padded pow2 for bitonic
#define MIN_SAMPLES 200

// ---------------------------------------------------------------------------
// 1) ordered compaction: first 64 indices of each class + total class count
// ---------------------------------------------------------------------------
__global__ void k_select(const int* __restrict__ labels, int n,
                         int* __restrict__ idx, int* __restrict__ cnt)
{
    const int wid  = threadIdx.x >> 5;
    const int lane = threadIdx.x & 31;
    if (wid >= N_CLS) return;
    const int c = wid;
    int kept = 0;
    for (int base = 0; base < n; base += 32) {
        int i = base + lane;
        bool p = (i < n) && (labels[i] == c);
        unsigned mask = (unsigned)__ballot(p);   // wave32: low 32 bits
        if (p) {
            int pos = kept + __popc(mask & ((1u << lane) - 1u));
            if (pos < NP) idx[c * NP + pos] = i;
        }
        kept += __popc(mask);
    }
    if (lane == 0) cnt[c] = kept;
}

// ---------------------------------------------------------------------------
// 2) row norms ||x_m||^2 for the 64 gathered points per class
// ---------------------------------------------------------------------------
__global__ void k_norms(const float* __restrict__ X, const int* __restrict__ idx,
                        float* __restrict__ norms)
{
    const int c = blockIdx.x, m = threadIdx.x;
    const float* row = X + (size_t)idx[c * NP + m] * DIMF;
    float s = 0.f;
    for (int k = 0; k < DIMF; ++k) { float v = row[k]; s = fmaf(v, v, s); }
    norms[c * NP + m] = s;
}

// ---------------------------------------------------------------------------
// 3) Gram via V_WMMA_F32_16X16X4_F32 (full fp32 — distance ORDER must match
//    the reference, so no bf16/fp8 shortcuts), then D = sqrt(clip(...))
//    One wave per 16x16 tile: 48 blocks x 32 threads. EXEC all-ones.
// ---------------------------------------------------------------------------
__global__ void k_gram(const float* __restrict__ X, const int* __restrict__ idx,
                       const float* __restrict__ norms, float* __restrict__ Dm)
{
    const int c    = blockIdx.x >> 4;
    const int tile = blockIdx.x & 15;
    const int tm = tile >> 2, tn = tile & 3;
    const int lane = threadIdx.x;
    const int half = lane >> 4;     // 0: K=0,1   1: K=2,3   (A 16x4 f32 layout)
    const int m    = lane & 15;

    const float* rowPtr = X + (size_t)idx[c * NP + tm * 16 + m] * DIMF;
    const float* colPtr = X + (size_t)idx[c * NP + tn * 16 + m] * DIMF;

    v8f acc = {0.f, 0.f, 0.f, 0.f, 0.f, 0.f, 0.f, 0.f};
    for (int k = 0; k < DIMF; k += 4) {
        const float2 av = *(const float2*)(rowPtr + k + 2 * half);
        const float2 bv = *(const float2*)(colPtr + k + 2 * half);
        v2f a, b;
        a[0] = av.x; a[1] = av.y;
        b[0] = bv.x; b[1] = bv.y;
#if defined(__gfx1250__)
#if __has_builtin(__builtin_amdgcn_wmma_f32_16x16x4_f32)
        acc = __builtin_amdgcn_wmma_f32_16x16x4_f32(
                  false, a, false, b, (short)0, acc, false, false);
#else
#error "gfx1250 device pass: f32 WMMA builtin not available"
#endif
#else
        // host pass / other arch: never executed; keep types satisfied
        acc[0] += a[0] * b[0];
#endif
    }

    // C/D layout: VGPR r, lane l -> M = r + (l>=16 ? 8 : 0), N = l & 15
    for (int r = 0; r < 8; ++r) {
        int gi = tm * 16 + r + 8 * half;
        int gj = tn * 16 + m;
        float d2 = norms[c * NP + gi] + norms[c * NP + gj] - 2.0f * acc[r];
        Dm[c * NP * NP + gi * NP + gj] = sqrtf(fmaxf(d2, 1e-12f));
    }
}

// ---------------------------------------------------------------------------
// 4) exact edge ranks: lexicographic (val, i, j), O(NE^2) counting sort
// ---------------------------------------------------------------------------
__global__ void k_edges(const float* __restrict__ Dm, int* __restrict__ rank,
                        float* __restrict__ ev)
{
    int g = blockIdx.x * blockDim.x + threadIdx.x;
    if (g >= N_CLS * NE) return;
    int c = g / NE, e = g % NE;
    int i = 0, rem = e;
    while (rem >= NP - 1 - i) { rem -= NP - 1 - i; ++i; }
    int j = i + 1 + rem;
    const float* D = Dm + c * NP * NP;
    float v = D[i * NP + j];
    int rk = 0;
    for (int a = 0; a < NP - 1; ++a)
        for (int b = a + 1; b < NP; ++b) {
            float w = D[a * NP + b];
            if (w < v || (w == v && (a < i || (a == i && b < j)))) ++rk;
        }
    rank[c * NP * NP + i * NP + j] = rk;
    rank[c * NP * NP + j * NP + i] = rk;
    ev[c * 2048 + rk] = v;
}

// ---------------------------------------------------------------------------
// 5) triangle sort keys: (maxr, r12, r02, r01) packed in one u64 (11b each)
// ---------------------------------------------------------------------------
__global__ void k_tri(const int* __restrict__ rank, unsigned long long* __restrict__ keys)
{
    int g = blockIdx.x * blockDim.x + threadIdx.x;
    if (g >= N_CLS * NSORT) return;
    int c = g / NSORT, t = g % NSORT;
    unsigned long long key = ~0ull;           // padding -> sorts to the end
    if (t < NT) {
        int a = 0, rem = t;
        for (;;) {
            int c2 = (NP - 1 - a) * (NP - 2 - a) / 2;
            if (rem < c2) break;
            rem -= c2; ++a;
        }
        int s = NP - 1 - a;
        int bp = 0;
        while (rem >= s - 1 - bp) { rem -= s - 1 - bp; ++bp; }
        int b  = a + 1 + bp;
        int cc = a + 2 + bp + rem;
        const int* R = rank + c * NP * NP;
        unsigned r01 = (unsigned)R[a * NP + b];
        unsigned r02 = (unsigned)R[a * NP + cc];
        unsigned r12 = (unsigned)R[b * NP + cc];
        unsigned mx = r01 > r02 ? r01 : r02; if (r12 > mx) mx = r12;
        key = ((unsigned long long)mx  << 33) | ((unsigned long long)r12 << 22)
            | ((unsigned long long)r02 << 11) |  (unsigned long long)r01;
    }
    keys[(size_t)c * NSORT + t] = key;
}

// ---------------------------------------------------------------------------
// 6) in-block bitonic sort of 65536 u64 keys (one workgroup per class)
// ---------------------------------------------------------------------------
__global__ void k_sort(unsigned long long* __restrict__ keysAll)
{
    unsigned long long* keys = keysAll + (size_t)blockIdx.x * NSORT;
    const unsigned tid = threadIdx.x, nth = blockDim.x;
    for (unsigned k = 2; k <= NSORT; k <<= 1) {
        for (unsigned j = k >> 1; j > 0; j >>= 1) {
            for (unsigned i = tid; i < NSORT; i += nth) {
                unsigned l = i ^ j;
                if (l > i) {
                    unsigned long long x = keys[i], y = keys[l];
                    bool asc = ((i & k) == 0);
                    if ((x > y) == asc) { keys[i] = y; keys[l] = x; }
                }
            }
            __threadfence_block();
            __syncthreads();
        }
    }
}

// ---------------------------------------------------------------------------
// 7) boundary-matrix reduction. One wave per class. Column = 2016-bit bitset
//    as 64 u32 words; lane L owns words L and L+32 of every column, so all
//    global colstore traffic is lane-private (no cross-lane coherence needed).
//    Pivot occupancy flags live in LDS. Pivot = wave-max via shfl_xor.
// ---------------------------------------------------------------------------
__global__ void k_reduce(const unsigned long long* __restrict__ keysAll,
                         const float* __restrict__ evAll,
                         unsigned int* __restrict__ colstoreAll,
                         float* __restrict__ tpOut)
{
    const int c    = blockIdx.x;
    const int lane = threadIdx.x;
    const unsigned long long* keys = keysAll + (size_t)c * NSORT;
    const float* ev = evAll + c * 2048;
    unsigned int* colstore = colstoreAll + (size_t)c * NE * 64;

    __shared__ unsigned char used[2048];
    for (int i = lane; i < 2048; i += 32) used[i] = 0;
    __syncthreads();

    float tp = 0.f;
    for (int t = 0; t < NSORT; ++t) {
        unsigned long long key = keys[t];
        if (key == ~0ull) break;               // padding: done
        unsigned r01 = (unsigned)( key        & 2047u);
        unsigned r02 = (unsigned)((key >> 11) & 2047u);
        unsigned r12 = (unsigned)((key >> 22) & 2047u);
        unsigned mx  = (unsigned)( key >> 33);

        unsigned w0 = 0, w1 = 0;
        {
            unsigned rrs[3] = {r01, r02, r12};
            for (int q = 0; q < 3; ++q) {
                int word = (int)(rrs[q] >> 5);
                unsigned bit = rrs[q] & 31u;
                if (word == lane)      w0 ^= (1u << bit);
                if (word == lane + 32) w1 ^= (1u << bit);
            }
        }
        for (;;) {
            int i1 = w1 ? (((lane + 32) << 5) + (31 - __clz((int)w1))) : -1;
            int i0 = w0 ? (((lane     ) << 5) + (31 - __clz((int)w0))) : -1;
            int local = i0 > i1 ? i0 : i1;
            for (int off = 16; off > 0; off >>= 1) {
                int o = __shfl_xor(local, off, 32);
                if (o > local) local = o;
            }
            if (local < 0) break;              // column reduced to zero
            int piv = local;
            if (!used[piv]) {
                used[piv] = 1;                 // uniform value, benign race
                colstore[(size_t)piv * 64 + lane]      = w0;
                colstore[(size_t)piv * 64 + lane + 32] = w1;
                tp += ev[mx] - ev[piv];        // death - birth
                break;
            } else {
                w0 ^= colstore[(size_t)piv * 64 + lane];
                w1 ^= colstore[(size_t)piv * 64 + lane + 32];
            }
        }
    }
    if (lane == 0) tpOut[c] = tp;
}

// ---------------------------------------------------------------------------
// 8) finalize: mean over valid classes of (tp - target_total_persistence)^2
// ---------------------------------------------------------------------------
__global__ void k_final(const float* __restrict__ tgt, const int* __restrict__ cnt,
                        const float* __restrict__ tp, float* __restrict__ out)
{
    if (threadIdx.x != 0 || blockIdx.x != 0) return;
    float total = 0.f; int valid = 0;
    for (int c = 0; c < N_CLS; ++c) {
        if (cnt[c] < MIN_SAMPLES) continue;
        float tpt = 0.f;
        for (int k = 0; k < 100; ++k)
            tpt += tgt[c * 200 + k * 2 + 1] - tgt[c * 200 + k * 2];
        float d = tp[c] - tpt;
        total += d * d;
        ++valid;
    }
    out[0] = valid ? total / (float)valid : 0.f;
}

// ---------------------------------------------------------------------------
extern "C" void kernel_launch(void* const* d_in, const int* in_sizes, int n_in,
                              void* d_out, int out_size, void* d_ws, size_t ws_size,
                              hipStream_t stream)
{
    (void)n_in; (void)out_size; (void)ws_size;
    const float* latent = (const float*)d_in[0];
    const int*   labels = (const int*)d_in[1];
    const float* tgt    = (const float*)d_in[2];
    const int n = in_sizes[1];

    // workspace layout (~3.1 MB total, fits comfortably in L2)
    char* ws = (char*)d_ws;
    int*                idx      = (int*)(ws + 0);        //  3*64 i32
    int*                cnt      = (int*)(ws + 768);      //  3    i32
    float*              norms    = (float*)(ws + 1024);   //  3*64 f32
    float*              tp       = (float*)(ws + 1792);   //  3    f32
    float*              Dm       = (float*)(ws + 2048);   //  3*4096 f32
    int*                rank     = (int*)(ws + 51200);    //  3*4096 i32
    float*              ev       = (float*)(ws + 100352); //  3*2048 f32
    unsigned long long* keys     = (unsigned long long*)(ws + 124928);  // 3*65536 u64
    unsigned int*       colstore = (unsigned int*)(ws + 1697792);       // 3*2016*64 u32

    k_select<<<1, 96, 0, stream>>>(labels, n, idx, cnt);
    k_norms <<<N_CLS, NP, 0, stream>>>(latent, idx, norms);
    k_gram  <<<N_CLS * 16, 32, 0, stream>>>(latent, idx, norms, Dm);
    k_edges <<<(N_CLS * NE + 255) / 256, 256, 0, stream>>>(Dm, rank, ev);
    k_tri   <<<(N_CLS * NSORT + 255) / 256, 256, 0, stream>>>(rank, keys);
    k_sort  <<<N_CLS, 1024, 0, stream>>>(keys);
    k_reduce<<<N_CLS, 32, 0, stream>>>(keys, ev, colstore, tp);
    k_final <<<1, 32, 0, stream>>>(tgt, cnt, tp, (float*)d_out);
}